// PtConv_23914377904591
// MI455X (gfx1250) — compile-verified
//
#include <hip/hip_runtime.h>

typedef __attribute__((ext_vector_type(2))) float v2f;
typedef __attribute__((ext_vector_type(4))) float v4f;
typedef __attribute__((ext_vector_type(8))) float v8f;

#define NB   8192
#define MB   8192
#define CIN  64
#define COUT 64
#define KS   16
#define KNB  16

static __device__ __forceinline__ v8f wmma4(v2f a, v2f b, v8f c) {
  // D = A(16x4,f32) * B(4x16,f32) + C(16x16,f32)
  return __builtin_amdgcn_wmma_f32_16x16x4_f32(false, a, false, b, (short)0, c,
                                               false, false);
}

// ---- one-shot: swizzle weight [1024x64] into WMMA B-fragment order ----
// e = kkg*256 + op*128 + lane*4 + q ; q = (ot&1)*2 + t ; ot = op*2 + (q>>1)
// wf[e] = W[(kkg*4 + (lane>>4)*2 + t)*64 + ot*16 + (lane&15)]
__global__ __launch_bounds__(256) void prep_wfrag(const float* __restrict__ w,
                                                  float* __restrict__ wf) {
  const int e    = blockIdx.x * 256 + threadIdx.x;   // 0..65535
  const int q    = e & 3;
  const int lane = (e >> 2) & 31;
  const int op   = (e >> 7) & 1;
  const int kkg  = e >> 8;
  const int t    = q & 1;
  const int ot   = op * 2 + (q >> 1);
  const int row  = kkg * 4 + ((lane >> 4) << 1) + t;
  const int col  = ot * 16 + (lane & 15);
  wf[e] = w[row * 64 + col];
}

template <bool USE_WF>
__global__ __launch_bounds__(32) void ptconv_wmma(
    const float* __restrict__ features,   // [B,N,CIN]
    const float* __restrict__ input_pts,  // [B,N,3]
    const float* __restrict__ output_pts, // [B,M,3]
    const int*   __restrict__ indices,    // [B,M,KNB]
    const float* __restrict__ weight,     // [CIN,KS,COUT] == Wflat[1024,64]
    const float* __restrict__ bias,       // [COUT]
    const float* __restrict__ centers,    // [3,KS]
    const float* __restrict__ l1w, const float* __restrict__ l1b, // [48,32],[32]
    const float* __restrict__ l2w, const float* __restrict__ l2b, // [32,16],[16]
    const float* __restrict__ l3w, const float* __restrict__ l3b, // [16,16],[16]
    const float* __restrict__ wfrag,      // pre-swizzled weight fragments
    float* __restrict__ out)              // [B,M,COUT]
{
  __shared__ float s_cent[48];
  __shared__ float s_l1wf[12 * 32 * 4];   // l1 B-frags: [kk][lane][q=nt*2+t]
  __shared__ float s_l1b[32];
  __shared__ float s_l2wf[8 * 32 * 2];    // l2 B-frags: [kk][lane][t]
  __shared__ float s_l2b[16];
  __shared__ float s_l3wf[4 * 32 * 2];    // l3 B-frags
  __shared__ float s_l3b[16];
  __shared__ int   s_idx[16 * 16];        // [p][n]
  __shared__ float s_rel[16 * 3 * 16];    // [p][dim][n]
  __shared__ float s_h1[16 * 34];         // [n][32col pad34] (8B-aligned pairs)
  __shared__ float s_h2[16 * 18];         // [n][16col pad18]
  __shared__ float s_hb[16 * 4 * 32 * 2]; // h in B-frag order: [p][kk][lane][t]
  __shared__ float s_g[16 * 260];         // [p][j(256) pad260]

  const int lane = threadIdx.x;       // full wave, EXEC all ones
  const int r16  = lane & 15;
  const int khi  = lane >> 4;
  const int gid  = blockIdx.x;
  const int b    = gid >> 9;
  const int m0   = (gid & 511) << 4;

  // ---- stage MLP weights directly in fragment order ----
  for (int i = lane; i < 48; i += 32) s_cent[i] = centers[i];
  s_l1b[lane] = l1b[lane];
  if (lane < 16) { s_l2b[lane] = l2b[lane]; s_l3b[lane] = l3b[lane]; }
#pragma unroll
  for (int kk = 0; kk < 12; ++kk)
#pragma unroll
    for (int q = 0; q < 4; ++q) {
      const int nt = q >> 1, t = q & 1;
      s_l1wf[(kk * 32 + lane) * 4 + q] =
          l1w[(kk * 4 + khi * 2 + t) * 32 + nt * 16 + r16];
    }
#pragma unroll
  for (int kk = 0; kk < 8; ++kk)
#pragma unroll
    for (int t = 0; t < 2; ++t)
      s_l2wf[(kk * 32 + lane) * 2 + t] = l2w[(kk * 4 + khi * 2 + t) * 16 + r16];
#pragma unroll
  for (int kk = 0; kk < 4; ++kk)
#pragma unroll
    for (int t = 0; t < 2; ++t)
      s_l3wf[(kk * 32 + lane) * 2 + t] = l3w[(kk * 4 + khi * 2 + t) * 16 + r16];

  // ---- gather neighbor indices + relative coordinates ----
  for (int q = lane; q < 256; q += 32) {
    const int p = q >> 4, n = q & 15;
    const int idx = indices[(b * MB + m0 + p) * KNB + n];
    s_idx[q] = idx;
    const float* ip = input_pts  + (size_t)(b * NB + idx) * 3;
    const float* op = output_pts + (size_t)(b * MB + m0 + p) * 3;
    s_rel[(p * 3 + 0) * 16 + n] = ip[0] - op[0];
    s_rel[(p * 3 + 1) * 16 + n] = ip[1] - op[1];
    s_rel[(p * 3 + 2) * 16 + n] = ip[2] - op[2];
  }
  __syncthreads();

  // ---- per-point MLP: d[16x48] -> h[16x16] (stored in B-frag order) ----
  for (int p = 0; p < 16; ++p) {
    v8f acc0 = {}; v8f acc1 = {};
#pragma unroll
    for (int kk = 0; kk < 12; ++kk) {
      const int c0  = kk * 4 + khi * 2;
      const int dim = kk >> 2;
      const int s0  = c0 & 15;
      const float rv = s_rel[(p * 3 + dim) * 16 + r16];
      v2f a;  a.x = rv - s_cent[dim * 16 + s0];
              a.y = rv - s_cent[dim * 16 + s0 + 1];
      const v4f wl = *(const v4f*)&s_l1wf[(kk * 32 + lane) * 4];
      const v2f b0 = __builtin_shufflevector(wl, wl, 0, 1);
      const v2f b1 = __builtin_shufflevector(wl, wl, 2, 3);
      acc0 = wmma4(a, b0, acc0);
      acc1 = wmma4(a, b1, acc1);
    }
    const float bb0 = s_l1b[r16], bb1 = s_l1b[16 + r16];
#pragma unroll
    for (int r = 0; r < 8; ++r) {
      const int n = r + khi * 8;
      s_h1[n * 34 + r16]      = fmaxf(acc0[r] + bb0, 0.0f);
      s_h1[n * 34 + 16 + r16] = fmaxf(acc1[r] + bb1, 0.0f);
    }
    __syncthreads();

    v8f acc2 = {};
#pragma unroll
    for (int kk = 0; kk < 8; ++kk) {
      const int c0 = kk * 4 + khi * 2;
      const v2f a  = *(const v2f*)&s_h1[r16 * 34 + c0];
      const v2f bq = *(const v2f*)&s_l2wf[(kk * 32 + lane) * 2];
      acc2 = wmma4(a, bq, acc2);
    }
    const float bb2 = s_l2b[r16];
#pragma unroll
    for (int r = 0; r < 8; ++r)
      s_h2[(r + khi * 8) * 18 + r16] = fmaxf(acc2[r] + bb2, 0.0f);
    __syncthreads();

    v8f acc3 = {};
#pragma unroll
    for (int kk = 0; kk < 4; ++kk) {
      const int c0 = kk * 4 + khi * 2;
      const v2f a  = *(const v2f*)&s_h2[r16 * 18 + c0];
      const v2f bq = *(const v2f*)&s_l3wf[(kk * 32 + lane) * 2];
      acc3 = wmma4(a, bq, acc3);
    }
    const float bb3 = s_l3b[r16];
    // store h[n][k] directly in B-fragment order:
    // addr = ((p*4 + n>>2)*32 + ((n>>1)&1)*16 + k)*2 + (n&1)
#pragma unroll
    for (int r = 0; r < 8; ++r) {
      const int n = r + khi * 8;
      s_hb[((p * 4 + (n >> 2)) * 32 + ((n >> 1) & 1) * 16 + r16) * 2 + (n & 1)] =
          fmaxf(acc3[r] + bb3, 0.0f);
    }
    __syncthreads();
  }

  // ---- G = f^T h per 16-channel block, then Out += Gstack @ Wblock ----
  const float* fbase = features + (size_t)b * NB * CIN;
  v8f oacc[4] = {v8f{}, v8f{}, v8f{}, v8f{}};

  for (int cb = 0; cb < 4; ++cb) {
    for (int p = 0; p < 16; ++p) {
      v8f g = {};
#pragma unroll
      for (int kk = 0; kk < 4; ++kk) {
        const int n0 = kk * 4 + khi * 2;
        v2f a;  // A[M=c=r16][K=n] = features[b, idx[p][n], cb*16 + c]
        a.x = fbase[(size_t)s_idx[p * 16 + n0]     * CIN + cb * 16 + r16];
        a.y = fbase[(size_t)s_idx[p * 16 + n0 + 1] * CIN + cb * 16 + r16];
        const v2f bq = *(const v2f*)&s_hb[((p * 4 + kk) * 32 + lane) * 2];
        g = wmma4(a, bq, g);
      }
#pragma unroll
      for (int r = 0; r < 8; ++r)
        s_g[p * 260 + (r + khi * 8) * 16 + r16] = g[r];
    }
    __syncthreads();

    if (USE_WF) {
      const float* wf_cb = wfrag + (size_t)cb * 64 * 256;
#pragma unroll 8
      for (int kk = 0; kk < 64; ++kk) {
        const int j0 = kk * 4 + khi * 2;
        const v2f a = *(const v2f*)&s_g[r16 * 260 + j0];
#pragma unroll
        for (int op = 0; op < 2; ++op) {
          const v4f wv = *(const v4f*)&wf_cb[(kk * 2 + op) * 128 + lane * 4];
          const v2f b0 = __builtin_shufflevector(wv, wv, 0, 1);
          const v2f b1 = __builtin_shufflevector(wv, wv, 2, 3);
          oacc[op * 2]     = wmma4(a, b0, oacc[op * 2]);
          oacc[op * 2 + 1] = wmma4(a, b1, oacc[op * 2 + 1]);
        }
      }
    } else {
      const float* wbase = weight + (size_t)cb * 256 * COUT;
#pragma unroll 4
      for (int kk = 0; kk < 64; ++kk) {
        const int j0 = kk * 4 + khi * 2;
        const v2f a = *(const v2f*)&s_g[r16 * 260 + j0];
#pragma unroll
        for (int ot = 0; ot < 4; ++ot) {
          v2f bq;
          bq.x = wbase[(size_t)j0       * COUT + ot * 16 + r16];
          bq.y = wbase[(size_t)(j0 + 1) * COUT + ot * 16 + r16];
          oacc[ot] = wmma4(a, bq, oacc[ot]);
        }
      }
    }
    __syncthreads();
  }

  // ---- epilogue: /Knb + bias, store ----
  const float inv = 1.0f / 16.0f;
#pragma unroll
  for (int ot = 0; ot < 4; ++ot) {
    const float bb = bias[ot * 16 + r16];
#pragma unroll
    for (int r = 0; r < 8; ++r) {
      const int p = r + khi * 8;
      out[(size_t)(b * MB + m0 + p) * COUT + ot * 16 + r16] =
          oacc[ot][r] * inv + bb;
    }
  }
}

extern "C" void kernel_launch(void* const* d_in, const int* in_sizes, int n_in,
                              void* d_out, int out_size, void* d_ws, size_t ws_size,
                              hipStream_t stream) {
  const float* features   = (const float*)d_in[0];
  const float* input_pts  = (const float*)d_in[1];
  const float* output_pts = (const float*)d_in[2];
  const int*   indices    = (const int*)  d_in[3];
  const float* weight     = (const float*)d_in[4];
  const float* bias       = (const float*)d_in[5];
  const float* centers    = (const float*)d_in[6];
  const float* l1w        = (const float*)d_in[7];
  const float* l1b        = (const float*)d_in[8];
  const float* l2w        = (const float*)d_in[9];
  const float* l2b        = (const float*)d_in[10];
  const float* l3w        = (const float*)d_in[11];
  const float* l3b        = (const float*)d_in[12];
  float* out = (float*)d_out;

  const dim3 grid(8 * (MB / 16));   // one 1-wave block per 16 output points
  const size_t wf_bytes = 1024 * 64 * sizeof(float);   // 256 KB

  if (ws_size >= wf_bytes) {
    float* wfrag = (float*)d_ws;
    prep_wfrag<<<256, 256, 0, stream>>>(weight, wfrag);
    ptconv_wmma<true><<<grid, 32, 0, stream>>>(features, input_pts, output_pts,
                                               indices, weight, bias, centers,
                                               l1w, l1b, l2w, l2b, l3w, l3b,
                                               wfrag, out);
  } else {
    ptconv_wmma<false><<<grid, 32, 0, stream>>>(features, input_pts, output_pts,
                                                indices, weight, bias, centers,
                                                l1w, l1b, l2w, l2b, l3w, l3b,
                                                (const float*)nullptr, out);
  }
}